// GAT_4269197492793
// MI455X (gfx1250) — compile-verified
//
#include <hip/hip_runtime.h>
#include <math.h>

// ---- problem constants (from the reference) ----
#define NN    50000
#define EE    400000
#define ETOT  450000          // EE + NN self-loops
#define F0    256             // IN_DIM
#define F1    512             // HEADS*HID
#define HEADS 8
#define OUTD  40
#define SLOPE 0.2f

typedef __attribute__((ext_vector_type(2)))  float    v2f;
typedef __attribute__((ext_vector_type(8)))  float    v8f;

__device__ __forceinline__ float leaky(float x) { return x > 0.f ? x : SLOPE * x; }

// float atomic max via integer punning (valid for mixed signs, -inf init)
__device__ __forceinline__ void atomicMaxF(float* a, float v) {
  if (v >= 0.f) atomicMax((int*)a, __float_as_int(v));
  else          atomicMin((unsigned int*)a, __float_as_uint(v));
}

__device__ __forceinline__ void edge_sd(const int* __restrict__ ei, int e, int& s, int& d) {
  if (e < EE) { s = ei[e]; d = ei[EE + e]; }
  else        { s = e - EE; d = e - EE; }      // appended self-loops
}

// -------------------- utility --------------------
__global__ void fill_kernel(float* __restrict__ p, float v, long n) {
  long i = (long)blockIdx.x * blockDim.x + threadIdx.x;
  if (i < n) p[i] = v;
}

// -------------------- WMMA GEMM --------------------
// C[M,Nc] = A[M,K] @ B[K,Nc], row-major f32.  One wave32 per 16x(16*NT) strip:
// one shared A fragment feeds NT back-to-back v_wmma_f32_16x16x4_f32 per K-step.
// BOUNDED=false: no predicates at all (used when Nc % (16*NT) == 0 and M % 16 == 0).
// BOUNDED=true : column clamp + v_cndmask select (no exec-masked loads).
template<int NT, bool BOUNDED>
__global__ void gemm_wmma(const float* __restrict__ A, const float* __restrict__ B,
                          float* __restrict__ C, int M, int Nc, int K) {
  int wave = (int)(((long)blockIdx.x * blockDim.x + threadIdx.x) >> 5);
  int lane = threadIdx.x & 31;
  int ngrp = (Nc + 16 * NT - 1) / (16 * NT);
  int tm = wave / ngrp;
  int tg = wave - tm * ngrp;
  if (tm * 16 >= M) return;                      // wave-uniform exit

  int mrow = tm * 16 + (lane & 15);
  int khi  = (lane >> 4) * 2;                    // A/B frag: lanes 0-15 K=+0/+1, 16-31 K=+2/+3
  const float* Arow = A + (size_t)mrow * K;

  int  ncol[NT];
  int  ncl[NT];                                  // clamped column for safe loads
  bool nok[NT];
  v8f  acc[NT];
#pragma unroll
  for (int t = 0; t < NT; ++t) {
    ncol[t] = (tg * NT + t) * 16 + (lane & 15);
    nok[t]  = !BOUNDED || (ncol[t] < Nc);
    ncl[t]  = BOUNDED ? (ncol[t] < Nc ? ncol[t] : Nc - 1) : ncol[t];
    acc[t]  = (v8f){};
  }

  for (int kk = 0; kk < K; kk += 4) {
    v2f a;
    a.x = Arow[kk + khi + 0];
    a.y = Arow[kk + khi + 1];
    const float* Br0 = B + (size_t)(kk + khi + 0) * Nc;
    const float* Br1 = B + (size_t)(kk + khi + 1) * Nc;
#pragma unroll
    for (int t = 0; t < NT; ++t) {
      v2f b;
      float bx = Br0[ncl[t]];
      float by = Br1[ncl[t]];
      b.x = (!BOUNDED || nok[t]) ? bx : 0.f;     // cndmask, not exec-masked load
      b.y = (!BOUNDED || nok[t]) ? by : 0.f;
      acc[t] = __builtin_amdgcn_wmma_f32_16x16x4_f32(false, a, false, b,
                                                     (short)0, acc[t], false, false);
    }
  }

  // C/D layout: VGPR r -> rows tm*16+r (lanes 0-15), tm*16+r+8 (lanes 16-31)
  int rbase = tm * 16 + (lane >> 4) * 8;
#pragma unroll
  for (int t = 0; t < NT; ++t) {
    if (!BOUNDED || nok[t]) {
#pragma unroll
      for (int r = 0; r < 8; ++r) {
        int row = rbase + r;
        if (!BOUNDED || row < M) C[(size_t)row * Nc + ncol[t]] = acc[t][r];
      }
    }
  }
}

// -------------------- per-node attention logits --------------------
// Layer 1: one wave per node; e_src[n,h] = sum_c h[n,h*64+c]*a_src[h,c]
__global__ void logits1_kernel(const float* __restrict__ h,
                               const float* __restrict__ a_src,
                               const float* __restrict__ a_dst,
                               float* __restrict__ es, float* __restrict__ ed) {
  int node = (int)(((long)blockIdx.x * blockDim.x + threadIdx.x) >> 5);
  int lane = threadIdx.x & 31;
  if (node >= NN) return;
  const float* row = h + (size_t)node * F1;
  float as[HEADS], ad[HEADS];
#pragma unroll
  for (int hh = 0; hh < HEADS; ++hh) {
    int c0 = hh * 64 + lane, c1 = c0 + 32;
    float v0 = row[c0], v1 = row[c1];
    as[hh] = v0 * a_src[c0] + v1 * a_src[c1];
    ad[hh] = v0 * a_dst[c0] + v1 * a_dst[c1];
  }
#pragma unroll
  for (int off = 16; off; off >>= 1)
#pragma unroll
    for (int hh = 0; hh < HEADS; ++hh) {
      as[hh] += __shfl_xor(as[hh], off, 32);
      ad[hh] += __shfl_xor(ad[hh], off, 32);
    }
  if (lane == 0)
#pragma unroll
    for (int hh = 0; hh < HEADS; ++hh) {
      es[node * HEADS + hh] = as[hh];
      ed[node * HEADS + hh] = ad[hh];
    }
}

// Layer 2: H=1, C=40
__global__ void logits2_kernel(const float* __restrict__ h,
                               const float* __restrict__ a_src,
                               const float* __restrict__ a_dst,
                               float* __restrict__ es, float* __restrict__ ed) {
  int node = (int)(((long)blockIdx.x * blockDim.x + threadIdx.x) >> 5);
  int lane = threadIdx.x & 31;
  if (node >= NN) return;
  const float* row = h + (size_t)node * OUTD;
  float v0 = row[lane];
  float v1 = (lane < 8) ? row[32 + lane] : 0.f;
  float a1 = (lane < 8) ? a_src[32 + lane] : 0.f;
  float b1 = (lane < 8) ? a_dst[32 + lane] : 0.f;
  float as = v0 * a_src[lane] + v1 * a1;
  float ad = v0 * a_dst[lane] + v1 * b1;
#pragma unroll
  for (int off = 16; off; off >>= 1) {
    as += __shfl_xor(as, off, 32);
    ad += __shfl_xor(ad, off, 32);
  }
  if (lane == 0) { es[node] = as; ed[node] = ad; }
}

// -------------------- segment softmax (3 passes) --------------------
__global__ void edge1_max(const int* __restrict__ ei, const float* __restrict__ es,
                          const float* __restrict__ ed, float* __restrict__ am) {
  long idx = (long)blockIdx.x * blockDim.x + threadIdx.x;
  if (idx >= (long)ETOT * HEADS) return;
  int e = (int)(idx >> 3), hh = (int)(idx & 7);
  int s, d; edge_sd(ei, e, s, d);
  atomicMaxF(&am[d * HEADS + hh], leaky(es[s * HEADS + hh] + ed[d * HEADS + hh]));
}

__global__ void edge1_sum(const int* __restrict__ ei, const float* __restrict__ es,
                          const float* __restrict__ ed, const float* __restrict__ am,
                          float* __restrict__ dn) {
  long idx = (long)blockIdx.x * blockDim.x + threadIdx.x;
  if (idx >= (long)ETOT * HEADS) return;
  int e = (int)(idx >> 3), hh = (int)(idx & 7);
  int s, d; edge_sd(ei, e, s, d);
  float a = leaky(es[s * HEADS + hh] + ed[d * HEADS + hh]);
  atomicAdd(&dn[d * HEADS + hh], __expf(a - am[d * HEADS + hh]));
}

// scatter: one wave32 per edge; channel c = 32k+lane, head = k>>1 (uniform per step)
__global__ void scatter1(const int* __restrict__ ei, const float* __restrict__ h,
                         const float* __restrict__ es, const float* __restrict__ ed,
                         const float* __restrict__ am, const float* __restrict__ dn,
                         float* __restrict__ out) {
  int e    = (int)(((long)blockIdx.x * blockDim.x + threadIdx.x) >> 5);
  int lane = threadIdx.x & 31;
  if (e >= ETOT) return;
  int s, d; edge_sd(ei, e, s, d);
  float coef = 0.f;
  if (lane < HEADS) {
    float a = leaky(es[s * HEADS + lane] + ed[d * HEADS + lane]);
    coef = __expf(a - am[d * HEADS + lane]) / dn[d * HEADS + lane];
  }
  const float* hrow = h   + (size_t)s * F1;
  float*       orow = out + (size_t)d * F1;
#pragma unroll
  for (int k = 0; k < 16; ++k) {
    float ck = __shfl(coef, k >> 1, 32);
    int c = k * 32 + lane;
    atomicAdd(&orow[c], hrow[c] * ck);
  }
}

__global__ void edge2_max(const int* __restrict__ ei, const float* __restrict__ es,
                          const float* __restrict__ ed, float* __restrict__ am) {
  long e = (long)blockIdx.x * blockDim.x + threadIdx.x;
  if (e >= ETOT) return;
  int s, d; edge_sd(ei, (int)e, s, d);
  atomicMaxF(&am[d], leaky(es[s] + ed[d]));
}

__global__ void edge2_sum(const int* __restrict__ ei, const float* __restrict__ es,
                          const float* __restrict__ ed, const float* __restrict__ am,
                          float* __restrict__ dn) {
  long e = (long)blockIdx.x * blockDim.x + threadIdx.x;
  if (e >= ETOT) return;
  int s, d; edge_sd(ei, (int)e, s, d);
  atomicAdd(&dn[d], __expf(leaky(es[s] + ed[d]) - am[d]));
}

__global__ void scatter2(const int* __restrict__ ei, const float* __restrict__ h,
                         const float* __restrict__ es, const float* __restrict__ ed,
                         const float* __restrict__ am, const float* __restrict__ dn,
                         float* __restrict__ out) {
  int e    = (int)(((long)blockIdx.x * blockDim.x + threadIdx.x) >> 5);
  int lane = threadIdx.x & 31;
  if (e >= ETOT) return;
  int s, d; edge_sd(ei, e, s, d);
  float coef = 0.f;
  if (lane == 0)
    coef = __expf(leaky(es[s] + ed[d]) - am[d]) / dn[d];
  coef = __shfl(coef, 0, 32);
  const float* hrow = h   + (size_t)s * OUTD;
  float*       orow = out + (size_t)d * OUTD;
  atomicAdd(&orow[lane], hrow[lane] * coef);
  if (lane < OUTD - 32)
    atomicAdd(&orow[32 + lane], hrow[32 + lane] * coef);
}

// -------------------- epilogues --------------------
__global__ void bias_elu_kernel(float* __restrict__ h, const float* __restrict__ b) {
  long i = (long)blockIdx.x * blockDim.x + threadIdx.x;
  if (i >= (long)NN * F1) return;
  float v = h[i] + b[i & (F1 - 1)];
  h[i] = v > 0.f ? v : (__expf(v) - 1.f);
}

__global__ void bias_out_kernel(float* __restrict__ o, const float* __restrict__ b) {
  long i = (long)blockIdx.x * blockDim.x + threadIdx.x;
  if (i >= (long)NN * OUTD) return;
  o[i] += b[(int)(i % OUTD)];
}

// -------------------- launcher --------------------
extern "C" void kernel_launch(void* const* d_in, const int* in_sizes, int n_in,
                              void* d_out, int out_size, void* d_ws, size_t ws_size,
                              hipStream_t stream) {
  const float* x     = (const float*)d_in[0];
  const int*   ei    = (const int*)  d_in[1];   // [2,E] flat: src=ei[0:E], dst=ei[E:2E]
  const float* W1    = (const float*)d_in[2];
  const float* asrc1 = (const float*)d_in[3];
  const float* adst1 = (const float*)d_in[4];
  const float* b1    = (const float*)d_in[5];
  const float* W2    = (const float*)d_in[6];
  const float* asrc2 = (const float*)d_in[7];
  const float* adst2 = (const float*)d_in[8];
  const float* b2    = (const float*)d_in[9];
  float* out = (float*)d_out;

  // workspace layout (~220 MB of floats)
  float* f    = (float*)d_ws;
  float* h1   = f;  f += (size_t)NN * F1;
  float* out1 = f;  f += (size_t)NN * F1;
  float* es1  = f;  f += (size_t)NN * HEADS;
  float* ed1  = f;  f += (size_t)NN * HEADS;
  float* am1  = f;  f += (size_t)NN * HEADS;
  float* dn1  = f;  f += (size_t)NN * HEADS;
  float* h2   = f;  f += (size_t)NN * OUTD;
  float* es2  = f;  f += (size_t)NN;
  float* ed2  = f;  f += (size_t)NN;
  float* am2  = f;  f += (size_t)NN;
  float* dn2  = f;  f += (size_t)NN;

  const int T = 256;
  auto cdiv = [](long a, long b) { return (unsigned)((a + b - 1) / b); };
  const float NEG_INF = -__builtin_huge_valf();

  // init accumulators (harness poisons ws/out with 0xAA)
  fill_kernel<<<cdiv((long)NN * F1, T),   T, 0, stream>>>(out1, 0.f,     (long)NN * F1);
  fill_kernel<<<cdiv((long)NN * HEADS, T),T, 0, stream>>>(am1,  NEG_INF, (long)NN * HEADS);
  fill_kernel<<<cdiv((long)NN * HEADS, T),T, 0, stream>>>(dn1,  0.f,     (long)NN * HEADS);
  fill_kernel<<<cdiv((long)NN, T),        T, 0, stream>>>(am2,  NEG_INF, (long)NN);
  fill_kernel<<<cdiv((long)NN, T),        T, 0, stream>>>(dn2,  0.f,     (long)NN);
  fill_kernel<<<cdiv((long)NN * OUTD, T), T, 0, stream>>>(out,  0.f,     (long)NN * OUTD);

  // ---- layer 1 ----
  {
    // 50000%16==0 and 512%(16*4)==0 -> fully unbounded fast path
    long waves = ((long)NN / 16) * (F1 / 64);                 // 3125 * 8
    gemm_wmma<4, false><<<cdiv(waves * 32, T), T, 0, stream>>>(x, W1, h1, NN, F1, F0);
  }
  logits1_kernel<<<cdiv((long)NN * 32, T), T, 0, stream>>>(h1, asrc1, adst1, es1, ed1);
  edge1_max<<<cdiv((long)ETOT * HEADS, T), T, 0, stream>>>(ei, es1, ed1, am1);
  edge1_sum<<<cdiv((long)ETOT * HEADS, T), T, 0, stream>>>(ei, es1, ed1, am1, dn1);
  scatter1<<<cdiv((long)ETOT * 32, T), T, 0, stream>>>(ei, h1, es1, ed1, am1, dn1, out1);
  bias_elu_kernel<<<cdiv((long)NN * F1, T), T, 0, stream>>>(out1, b1);

  // ---- layer 2 ----
  {
    // Nc=40 ragged -> bounded variant (clamped loads + cndmask, masked stores)
    long waves = ((long)NN / 16) * 1;                         // 3125 strips of 16x64
    gemm_wmma<4, true><<<cdiv(waves * 32, T), T, 0, stream>>>(out1, W2, h2, NN, OUTD, F1);
  }
  logits2_kernel<<<cdiv((long)NN * 32, T), T, 0, stream>>>(h2, asrc2, adst2, es2, ed2);
  edge2_max<<<cdiv((long)ETOT, T), T, 0, stream>>>(ei, es2, ed2, am2);
  edge2_sum<<<cdiv((long)ETOT, T), T, 0, stream>>>(ei, es2, ed2, am2, dn2);
  scatter2<<<cdiv((long)ETOT * 32, T), T, 0, stream>>>(ei, h2, es2, ed2, am2, dn2, out);
  bias_out_kernel<<<cdiv((long)NN * OUTD, T), T, 0, stream>>>(out, b2);
}